// MemristorDense_85787676771025
// MI455X (gfx1250) — compile-verified
//
#include <hip/hip_runtime.h>
#include <hip/hip_bf16.h>

// ---------------------------------------------------------------------------
// Memristor crossbar dense forward, reformulated for MI455X (gfx1250).
//
//   I[b,c] = sum_r Gv[r,c] * s[b,r] * exp(lr[b,r] * gamma[r,c])
//
// gamma = log2(n), n ~ N(2.132, 0.095)  =>  gamma = GAMMA0 + delta, |delta| small.
// Taylor in delta (8 terms) turns the trans-bound 134M-exp tensor op into ONE
// f32 GEMM  [128 x 8256] x [8256 x 512]  on V_WMMA_F32_16X16X4_F32, plus ~2M
// transcendentals to build the factors.  pos-neg column subtraction is folded
// into the B factor (N = 512 instead of 1024).
//
// GEMM inner loop: B stored pair-major and column-permuted:
//   B[k][w*64+t*16+c]  ->  Bpack[((k/2)*512 + w*64 + c*4 + t)*2 + (k&1)]
// so each lane's 4-K-step fragments are 8 consecutive floats = two b128 loads
// whose aligned dword pairs ARE the v2f WMMA operands (no repack movs).
// M supertile of 32 rows/wave => 4 VMEM instructions per 8 WMMAs.
// ---------------------------------------------------------------------------

typedef __attribute__((ext_vector_type(2))) float v2f;
typedef __attribute__((ext_vector_type(4))) float v4f;
typedef __attribute__((ext_vector_type(8))) float v8f;

#define BATCH   128
#define NIN     1024
#define NOUT    512
#define R_REAL  1025          // n_in + bias row
#define R_PAD   1032          // padded (even) so KHAT/NKCH is a multiple of 4
#define KTERMS  8             // Taylor terms (orders 0..7)
#define KHAT    (KTERMS * R_PAD)   // 8256, GEMM inner dimension
#define NKCH    16            // K split across blocks
#define KCHUNK  (KHAT / NKCH) // 516, multiple of 4
#define MSUP    4             // M supertiles of 32 rows

#define V_REF   0.25f
#define K_V     0.5f          // 2 * V_REF
#define G_OFF   1e-4f
#define G_ON    1e-3f
#define GAMMA0  1.0922335f    // log2(2.132)

// ---- kernel 0: zero output + ws scalars ----------------------------------
__global__ void mk_zero(float* __restrict__ out, unsigned* __restrict__ scal_u) {
  int idx = blockIdx.x * blockDim.x + threadIdx.x;
  if (idx < BATCH * NOUT) out[idx] = 0.0f;
  if (idx == 0) { scal_u[0] = 0u; scal_u[1] = 0u; scal_u[2] = 0u; }
}

// ---- kernel 1: max over combined weights (all values are >= 0) ------------
__global__ void mk_maxw(const float* __restrict__ wp, const float* __restrict__ wn,
                        const float* __restrict__ bp, const float* __restrict__ bn,
                        unsigned* __restrict__ umax) {
  __shared__ float smax[256];
  float v = 0.0f;
  const int total = NIN * NOUT;
  for (int i = blockIdx.x * blockDim.x + threadIdx.x; i < total;
       i += gridDim.x * blockDim.x)
    v = fmaxf(v, fmaxf(wp[i], wn[i]));
  if (blockIdx.x == 0)
    for (int i = threadIdx.x; i < NOUT; i += blockDim.x)
      v = fmaxf(v, fmaxf(bp[i], bn[i]));
  smax[threadIdx.x] = v;
  __syncthreads();
  for (int s = 128; s > 0; s >>= 1) {
    if ((int)threadIdx.x < s)
      smax[threadIdx.x] = fmaxf(smax[threadIdx.x], smax[threadIdx.x + s]);
    __syncthreads();
  }
  if (threadIdx.x == 0)
    atomicMax(umax, __float_as_uint(smax[0]));  // valid order for floats >= 0
}

// ---- kernel 2: scalars k_G and 1/(K_V*k_G) --------------------------------
__global__ void mk_scal(float* __restrict__ scal) {
  float max_w = __uint_as_float(((unsigned*)scal)[0]);
  float k_G = (G_ON - G_OFF) / max_w;
  scal[1] = k_G;
  scal[2] = 1.0f / (K_V * k_G);
}

// ---- kernel 3: build A-hat [BATCH x KHAT], A_k[b,r]=s*e^{lr*g0}*lr^k/k! ---
__global__ void mk_buildA(const float* __restrict__ x, float* __restrict__ Ah) {
  int idx = blockIdx.x * blockDim.x + threadIdx.x;
  if (idx >= BATCH * R_PAD) return;
  int b = idx / R_PAD;
  int r = idx % R_PAD;
  float s = 0.0f, lr = 0.0f;
  if (r < NIN) {
    float xv = x[(size_t)b * NIN + r];
    s  = (xv > 0.0f) ? 1.0f : ((xv < 0.0f) ? -1.0f : 0.0f);
    lr = __logf(fmaxf(2.0f * fabsf(xv), 1e-12f));   // ln(|V|/V_REF)
  } else if (r == NIN) {                            // bias row: V = K_V, ratio = 1
    s = 1.0f; lr = 0.0f;
  }                                                 // r >= 1025: zero padding
  float term = s * __expf(lr * GAMMA0);
  float* dst = Ah + (size_t)b * KHAT + r;
  dst[0] = term;
  for (int k = 1; k < KTERMS; ++k) {
    term = term * lr / (float)k;
    dst[(size_t)k * R_PAD] = term;
  }
}

// ---- kernel 4: build B-hat, pair-major + column-permuted ------------------
// khat = k*R_PAD + r (R_PAD even => parity = r&1):
//   Bpack[((khat>>1)*NOUT + jp)*2 + (r&1)] = Gp*dp^k - Gn*dn^k
// jp: j = w*64 + t*16 + c  ->  w*64 + c*4 + t
__global__ void mk_buildB(const float* __restrict__ wp, const float* __restrict__ wn,
                          const float* __restrict__ bp, const float* __restrict__ bn,
                          const float* __restrict__ np, const float* __restrict__ scal,
                          float* __restrict__ Bh) {
  int idx = blockIdx.x * blockDim.x + threadIdx.x;
  if (idx >= R_PAD * NOUT) return;
  int r = idx / NOUT;
  int j = idx % NOUT;
  float Gp = 0.0f, Gn = 0.0f, dp = 0.0f, dn = 0.0f;
  if (r < R_REAL) {
    float k_G = scal[1];
    float wpv = (r < NIN) ? wp[(size_t)r * NOUT + j] : bp[j];
    float wnv = (r < NIN) ? wn[(size_t)r * NOUT + j] : bn[j];
    Gp = (k_G * wpv + G_OFF) * V_REF;               // G * V_REF, pos column (2j)
    Gn = (k_G * wnv + G_OFF) * V_REF;               // neg column (2j+1)
    dp = __log2f(np[(size_t)r * (2 * NOUT) + 2 * j])     - GAMMA0;
    dn = __log2f(np[(size_t)r * (2 * NOUT) + 2 * j + 1]) - GAMMA0;
  }
  int jp  = (j & ~63) | ((j & 15) << 2) | ((j >> 4) & 3);
  int par = r & 1;
  float tp = Gp, tn = Gn;
  for (int k = 0; k < KTERMS; ++k) {
    if (k) { tp *= dp; tn *= dn; }
    size_t khat = (size_t)k * R_PAD + r;
    Bh[((khat >> 1) * NOUT + jp) * 2 + par] = tp - tn;
  }
}

// ---- kernel 5: f32 WMMA GEMM  y += inv_scale * Ah @ Bh --------------------
// Grid: 4 M-supertiles x 16 K-chunks; block = 256 (8 waves); wave owns 32x64.
__global__ __launch_bounds__(256) void mk_gemm(const float* __restrict__ Ah,
                                               const float* __restrict__ Bh,
                                               const float* __restrict__ scal,
                                               float* __restrict__ out) {
  const int msup   = blockIdx.x & 3;        // 4 supertiles of 32 rows
  const int kchunk = blockIdx.x >> 2;       // 16 chunks of 516
  const int wave   = threadIdx.x >> 5;
  const int lane   = threadIdx.x & 31;
  const int mbase  = msup * 32;
  const int nbase  = wave * 64;
  const int k0     = kchunk * KCHUNK;
  const int m      = lane & 15;
  const int khalf  = (lane >> 4) * 2;       // ISA A/B layout: lanes 16-31 hold K+2

  const float inv_scale = scal[2];

  v8f acc[2][4] = {};
  const float* Arow0 = Ah + (size_t)(mbase + m) * KHAT + k0 + khalf;
  const float* Arow1 = Arow0 + (size_t)16 * KHAT;
  // pair-major B: float offset = k2*(2*NOUT) + (jp)*2, k2 = (k0+khalf)/2 + kk/2
  const float* Bbase = Bh + (size_t)((k0 + khalf) >> 1) * (2 * NOUT)
                          + (size_t)(nbase + m * 4) * 2;

  for (int kk = 0; kk < KCHUNK; kk += 4) {
    // A fragments (16x4): lane m holds K = kk+khalf in v0, kk+khalf+1 in v1
    v2f a0; a0.x = Arow0[kk]; a0.y = Arow0[kk + 1];
    v2f a1; a1.x = Arow1[kk]; a1.y = Arow1[kk + 1];
    // B fragments: 8 consecutive floats -> two b128; dword pairs = v2f operands
    const float* brow = Bbase + (size_t)(kk >> 1) * (2 * NOUT);
    __builtin_prefetch(brow + 8 * NOUT, 0, 1);      // global_prefetch_b8, 16KB ahead
    v4f q0 = *(const v4f*)brow;                     // {t0 pair, t1 pair}
    v4f q1 = *(const v4f*)(brow + 4);               // {t2 pair, t3 pair}
    v2f b0 = {q0.x, q0.y};
    v2f b1 = {q0.z, q0.w};
    v2f b2 = {q1.x, q1.y};
    v2f b3 = {q1.z, q1.w};
    acc[0][0] = __builtin_amdgcn_wmma_f32_16x16x4_f32(false, a0, false, b0, (short)0, acc[0][0], false, false);
    acc[0][1] = __builtin_amdgcn_wmma_f32_16x16x4_f32(false, a0, false, b1, (short)0, acc[0][1], false, false);
    acc[0][2] = __builtin_amdgcn_wmma_f32_16x16x4_f32(false, a0, false, b2, (short)0, acc[0][2], false, false);
    acc[0][3] = __builtin_amdgcn_wmma_f32_16x16x4_f32(false, a0, false, b3, (short)0, acc[0][3], false, false);
    acc[1][0] = __builtin_amdgcn_wmma_f32_16x16x4_f32(false, a1, false, b0, (short)0, acc[1][0], false, false);
    acc[1][1] = __builtin_amdgcn_wmma_f32_16x16x4_f32(false, a1, false, b1, (short)0, acc[1][1], false, false);
    acc[1][2] = __builtin_amdgcn_wmma_f32_16x16x4_f32(false, a1, false, b2, (short)0, acc[1][2], false, false);
    acc[1][3] = __builtin_amdgcn_wmma_f32_16x16x4_f32(false, a1, false, b3, (short)0, acc[1][3], false, false);
  }

  // C/D layout: VGPR i -> row h*16 + i + 8*(lane>>4), col = lane&15 in tile t
  const int rbase = mbase + (lane >> 4) * 8;
  const int ncol  = lane & 15;
  for (int h = 0; h < 2; ++h)
    for (int t = 0; t < 4; ++t)
      for (int i = 0; i < 8; ++i) {
        float* o = out + (size_t)(rbase + h * 16 + i) * NOUT + nbase + t * 16 + ncol;
        atomicAdd(o, acc[h][t][i] * inv_scale);
      }
}

// ---------------------------------------------------------------------------
extern "C" void kernel_launch(void* const* d_in, const int* in_sizes, int n_in_args,
                              void* d_out, int out_size, void* d_ws, size_t ws_size,
                              hipStream_t stream) {
  const float* x  = (const float*)d_in[0];
  const float* wp = (const float*)d_in[1];
  const float* wn = (const float*)d_in[2];
  const float* bp = (const float*)d_in[3];
  const float* bn = (const float*)d_in[4];
  const float* np = (const float*)d_in[5];
  float* out = (float*)d_out;

  float*    scal = (float*)d_ws;                 // [0]=max bits, [1]=k_G, [2]=inv_scale
  float*    Ah   = scal + 64;                    // 128*8256 floats  (~4.2 MB)
  float*    Bh   = Ah + (size_t)BATCH * KHAT;    // 8256*512 floats  (~16.9 MB)

  mk_zero  <<<(BATCH * NOUT + 255) / 256, 256, 0, stream>>>(out, (unsigned*)scal);
  mk_maxw  <<<256, 256, 0, stream>>>(wp, wn, bp, bn, (unsigned*)scal);
  mk_scal  <<<1, 1, 0, stream>>>(scal);
  mk_buildA<<<(BATCH * R_PAD + 255) / 256, 256, 0, stream>>>(x, Ah);
  mk_buildB<<<(R_PAD * NOUT + 255) / 256, 256, 0, stream>>>(wp, wn, bp, bn, np, scal, Bh);
  mk_gemm  <<<MSUP * NKCH, 256, 0, stream>>>(Ah, Bh, scal, out);
}